// SegmentationShader_51917564674428
// MI455X (gfx1250) — compile-verified
//
#include <hip/hip_runtime.h>

// SegmentationShader: per-pixel-layer barycentric interpolation of per-vertex
// class scores, truncated to int32.
//
//   verts_class : [N*V]          f32   (D==1, packed view)
//   faces       : [F,3]          i32
//   pix_to_face : [N,H,W,K]      i32   (K==2)
//   bary_coords : [N,H,W,K,3]    f32
//   out         : [N,H,W,K]      i32
//
// Memory-bound gather (0.25 FLOP/byte). HBM floor ~14us @ 23.3 TB/s.
// CDNA5 specifics (verified in disasm):
//  * TH_LOAD_NT / TH_STORE_NT on the one-pass streams (p2f/bary/out) so the
//    L2-resident gather tables (faces 1.3MB, verts 220KB) stay hot.
//  * b64/b128 NT loads, b96 face-triple gather, scale_offset vertex gather.
//  * Branchless masking (cndmask instead of execz branches) — matches the
//    reference's where(mask,0,...) semantics and avoids exec-mask overhead
//    when valid/invalid faces mix within a wave.
//  * Exact grid (8,388,608 px / 256 = 32768 blocks): no tail guard.

#define N_  8
#define V_  6890
#define F_  110208
#define H_  1024
#define W_  1024
#define K_  2

// clang ext_vector types (required by __builtin_nontemporal_*); 8B -> b64 ops.
typedef int   v2i __attribute__((ext_vector_type(2)));
typedef float v2f __attribute__((ext_vector_type(2)));

__global__ __launch_bounds__(256) void seg_shader_kernel(
    const float* __restrict__ verts,   // [N*V]
    const int*   __restrict__ faces,   // [F*3]
    const int*   __restrict__ p2f,     // [N*H*W*K]
    const float* __restrict__ bary,    // [N*H*W*K*3]
    int*         __restrict__ out)     // [N*H*W*K]
{
    const int pix = blockIdx.x * blockDim.x + threadIdx.x;  // exact grid

    // Two face ids for this pixel (K=2): one NT b64 load.
    const v2i f2 = __builtin_nontemporal_load((const v2i*)p2f + pix);

    // Six barycentrics (24B, 8B aligned): NT b64 loads (compiler merges).
    const v2f* bptr = (const v2f*)(bary + (long long)pix * (K_ * 3));
    const v2f b01 = __builtin_nontemporal_load(bptr + 0); // k0: c0,c1
    const v2f b23 = __builtin_nontemporal_load(bptr + 1); // k0: c2 | k1: c0
    const v2f b45 = __builtin_nontemporal_load(bptr + 2); // k1: c1,c2

    // Branchless mask, exactly like the reference:
    //   idx = where(f<0, 0, f); gather; val = where(f<0, 0, val)
    const int f0 = f2.x;
    const int f1 = f2.y;
    const int g0 = (f0 < 0) ? 0 : f0;   // v_cndmask / v_max
    const int g1 = (f1 < 0) ? 0 : f1;

    // ---- layer k = 0 ---- (faces/verts: RT-temporal, L2-resident)
    const int a0 = faces[3 * g0 + 0];
    const int a1 = faces[3 * g0 + 1];
    const int a2 = faces[3 * g0 + 2];
    float v0 = fmaf(b01.x, verts[a0],
               fmaf(b01.y, verts[a1],
                    b23.x * verts[a2]));
    v0 = (f0 < 0) ? 0.0f : v0;

    // ---- layer k = 1 ----
    const int c0 = faces[3 * g1 + 0];
    const int c1 = faces[3 * g1 + 1];
    const int c2 = faces[3 * g1 + 2];
    float v1 = fmaf(b23.y, verts[c0],
               fmaf(b45.x, verts[c1],
                    b45.y * verts[c2]));
    v1 = (f1 < 0) ? 0.0f : v1;

    v2i res;
    res.x = (int)v0;   // C truncation == astype(jnp.int32)
    res.y = (int)v1;

    // One-pass output: NT b64 store, don't pollute caches.
    __builtin_nontemporal_store(res, (v2i*)out + pix);
}

extern "C" void kernel_launch(void* const* d_in, const int* in_sizes, int n_in,
                              void* d_out, int out_size, void* d_ws, size_t ws_size,
                              hipStream_t stream) {
    (void)in_sizes; (void)n_in; (void)out_size; (void)d_ws; (void)ws_size;

    const float* verts = (const float*)d_in[0];  // verts_class [N,V,1] -> [N*V]
    const int*   faces = (const int*)  d_in[1];  // faces       [F,3]
    const int*   p2f   = (const int*)  d_in[2];  // pix_to_face [N,H,W,K]
    const float* bary  = (const float*)d_in[3];  // bary_coords [N,H,W,K,3]
    int*         outp  = (int*)d_out;            // seg [N,H,W,K] (int32)

    const int n_pix   = N_ * H_ * W_;            // 8,388,608 pixels; K=2 fused
    const int threads = 256;                     // 8 wave32 waves per block
    const int blocks  = n_pix / threads;         // exact: 32768

    seg_shader_kernel<<<blocks, threads, 0, stream>>>(verts, faces, p2f, bary,
                                                      outp);
}